// KnnLoss_15762529976905
// MI455X (gfx1250) — compile-verified
//
#include <hip/hip_runtime.h>
#include <hip/hip_bf16.h>

typedef __attribute__((ext_vector_type(2))) float v2f;
typedef __attribute__((ext_vector_type(8))) float v8f;

#define NPTS   4096
#define BATCH  8
#define KNN    16
#define RAD2   0.0625f            // RADIUS^2, RADIUS = 0.25
#define WAVES  4                  // 128-thread block, wave32
#define QPERWG (WAVES * 32)       // 128 queries per workgroup

__global__ void KnnLoss_zero_out(float* out) { out[0] = 0.0f; }

__global__ __launch_bounds__(128) void KnnLoss_15762529976905_kernel(
    const float* __restrict__ pc, const float* __restrict__ flow,
    float* __restrict__ out) {
  __shared__ float pc_s[NPTS * 3];       // whole batch's points: 48 KB
  __shared__ float stage[WAVES][32][17]; // per-wave dot tile staging (pad 17)
  __shared__ float sqb_s[WAVES][16];     // candidate squared norms per tile
  __shared__ float s_sum;

  const int tid  = threadIdx.x;
  const int wave = tid >> 5;
  const int lane = tid & 31;
  const int b    = blockIdx.x >> 5;            // 32 workgroups per batch
  const int wgq0 = (blockIdx.x & 31) * QPERWG; // first query of this WG
  const int q    = wgq0 + wave * 32 + lane;    // this lane's query in [0,4096)

  if (tid == 0) s_sum = 0.0f;

  // Stage pc[b] into LDS (coalesced)
  const float* pcb = pc + (size_t)b * NPTS * 3;
  for (int i = tid; i < NPTS * 3; i += 128) pc_s[i] = pcb[i];
  __syncthreads();

  // Build A tiles (16x4 f32, K padded 3->4 with zero) per ISA VGPR layout:
  // VGPR0: lanes0-15 K=0, lanes16-31 K=2; VGPR1: lanes0-15 K=1, lanes16-31 K=3
  const int half = lane >> 4;
  const int m    = lane & 15;
  const int qlo  = wgq0 + wave * 32;  // lo tile rows = queries qlo..qlo+15
  const int qhi  = qlo + 16;          // hi tile rows = queries qlo+16..qlo+31
  v2f a_lo, a_hi;
  a_lo.x = pc_s[(qlo + m) * 3 + 2 * half];
  a_lo.y = half ? 0.0f : pc_s[(qlo + m) * 3 + 1];
  a_hi.x = pc_s[(qhi + m) * 3 + 2 * half];
  a_hi.y = half ? 0.0f : pc_s[(qhi + m) * 3 + 1];

  // Own query squared norm
  float qx = pc_s[q * 3 + 0], qy = pc_s[q * 3 + 1], qz = pc_s[q * 3 + 2];
  const float sq_q = qx * qx + qy * qy + qz * qz;

  // Sorted ascending top-K list in registers
  float bd[KNN];
  int   bi[KNN];
#pragma unroll
  for (int s = 0; s < KNN; ++s) { bd[s] = 3.4e38f; bi[s] = 0; }

  for (int j0 = 0; j0 < NPTS; j0 += 16) {
    // B tile (4x16 f32): same K striping as A across lane halves
    const int jn = j0 + m;
    v2f bb;
    bb.x = pc_s[jn * 3 + 2 * half];
    bb.y = half ? 0.0f : pc_s[jn * 3 + 1];
    if (half == 0) {
      float x = pc_s[jn * 3], y = pc_s[jn * 3 + 1], z = pc_s[jn * 3 + 2];
      sqb_s[wave][m] = x * x + y * y + z * z;
    }

    v8f c0 = {};
    v8f d_lo = __builtin_amdgcn_wmma_f32_16x16x4_f32(
        false, a_lo, false, bb, (short)0, c0, false, false);
    v8f d_hi = __builtin_amdgcn_wmma_f32_16x16x4_f32(
        false, a_hi, false, bb, (short)0, c0, false, false);

    // D layout: VGPR r, lanes0-15 -> (M=r, N=lane); lanes16-31 -> (M=r+8)
#pragma unroll
    for (int r = 0; r < 8; ++r) {
      stage[wave][r + half * 8][m]      = d_lo[r];
      stage[wave][16 + r + half * 8][m] = d_hi[r];
    }
    __syncthreads();

    // Lane owns query row `lane`: scan 16 candidates, guarded sorted insert
#pragma unroll
    for (int jj = 0; jj < 16; ++jj) {
      float dot = stage[wave][lane][jj];
      float d2  = sq_q + sqb_s[wave][jj] - 2.0f * dot;
      if (d2 < bd[KNN - 1]) {
        float v = d2; int vi = j0 + jj;
#pragma unroll
        for (int s = 0; s < KNN; ++s) {
          bool lt = v < bd[s];
          float tf = bd[s]; int ti = bi[s];
          bd[s] = lt ? v : bd[s];  bi[s] = lt ? vi : bi[s];
          v     = lt ? tf : v;     vi    = lt ? ti : vi;
        }
      }
    }
    __syncthreads();
  }

  // Loss: L1 flow diff vs 16 neighbors (out-of-radius -> nearest/self index)
  const float* fb = flow + (size_t)b * NPTS * 3;
  float fx = fb[q * 3 + 0], fy = fb[q * 3 + 1], fz = fb[q * 3 + 2];
  const int self = bi[0];
  float sum = 0.0f;
#pragma unroll
  for (int s = 0; s < KNN; ++s) {
    int nidx = (bd[s] > RAD2) ? self : bi[s];
    float nx = fb[nidx * 3 + 0], ny = fb[nidx * 3 + 1], nz = fb[nidx * 3 + 2];
    sum += fabsf(fx - nx) + fabsf(fy - ny) + fabsf(fz - nz);
  }

  atomicAdd(&s_sum, sum);
  __syncthreads();
  if (tid == 0)
    atomicAdd(out, s_sum * (1.0f / ((float)BATCH * NPTS * KNN)));
}

extern "C" void kernel_launch(void* const* d_in, const int* in_sizes, int n_in,
                              void* d_out, int out_size, void* d_ws, size_t ws_size,
                              hipStream_t stream) {
  const float* pc   = (const float*)d_in[0];
  const float* flow = (const float*)d_in[1];
  float* out = (float*)d_out;
  (void)in_sizes; (void)n_in; (void)out_size; (void)d_ws; (void)ws_size;

  KnnLoss_zero_out<<<dim3(1), dim3(1), 0, stream>>>(out);
  // 8 batches x 32 query-blocks of 128 queries; 128 threads = 4 wave32
  KnnLoss_15762529976905_kernel<<<dim3(BATCH * 32), dim3(128), 0, stream>>>(
      pc, flow, out);
}